// MultiheadAttention_21955872817209
// MI455X (gfx1250) — compile-verified
//
#include <hip/hip_runtime.h>

// ---------------------------------------------------------------------------
// MultiheadAttention forward for MI455X (gfx1250), fp32-faithful path using
// V_WMMA_F32_16X16X4_F32 + Tensor Data Mover (tensor_load_to_lds) staging.
// B=4, S=2048, E=512, H=8, d=64.
// ---------------------------------------------------------------------------

typedef __attribute__((ext_vector_type(2))) float v2f;
typedef __attribute__((ext_vector_type(8))) float v8f;
typedef unsigned int u32;
typedef __attribute__((ext_vector_type(4))) u32 u32x4;
typedef __attribute__((ext_vector_type(8))) u32 u32x8;

#define S_LEN 2048
#define EMB   512
#define NH    8
#define HD    64
#define BATCH 4

// D = A(16x4) * B(4x16) + C(16x16), all fp32.
// A layout: lane = M + 16*(K>=2), reg = K&1  -> regs hold consecutive K pair.
// B layout (mirror): lane = N + 16*(K>=2), reg = K&1.
// C/D layout: lane = N + 16*(M>=8), reg = M%8.
__device__ __forceinline__ v8f wmma4(v2f a, v2f b, v8f c) {
  return __builtin_amdgcn_wmma_f32_16x16x4_f32(
      /*neg_a=*/false, a, /*neg_b=*/false, b,
      /*c_mod=*/(short)0, c, /*reuse_a=*/false, /*reuse_b=*/false);
}

// ---------------------------------------------------------------------------
// TDM: load a 2-D fp32 tile (tile_w x tile_h elems, row stride `stride` elems)
// from global memory into LDS at byte offset lds_off, inserting 2 pad DWORDs
// after every 64 DWORDs (=> 66-float LDS rows, bank-conflict-free + 8B
// aligned for the WMMA fragment reads).  D# per CDNA5 ISA:
//   group0: count=1 | lds_addr | global_addr[56:0] | type=2
//   group1: data_size=4B, pad_enable, pad_interval=5 (64 dw), pad_amount=1
//           (2 dw), tensor_dim0/1 = tile dims, tile_dim0/1, dim0_stride.
// Issued once per wave (EXEC ignored by tensor ops); tracked on TENSORcnt.
// ---------------------------------------------------------------------------
__device__ __forceinline__ void tdm_load_tile_pad2(u32 lds_off,
                                                   const float* gptr,
                                                   u32 tile_w, u32 tile_h,
                                                   u32 stride_elems) {
  unsigned long long ga = (unsigned long long)(uintptr_t)gptr;
  u32x4 g0;
  g0[0] = 1u;                                           // count=1, user mode
  g0[1] = lds_off;                                      // LDS byte address
  g0[2] = (u32)ga;                                      // global_addr[31:0]
  g0[3] = (u32)((ga >> 32) & 0x01FFFFFFu) | (2u << 30); // addr[56:32] | type=2
  u32x8 g1;
  g1[0] = (2u << 16)            // data_size = 4 bytes
        | (1u << 20)            // pad_enable
        | (5u << 22)            // pad_interval: 64 DWORDs
        | (1u << 25);           // pad_amount:   2 DWORDs
  g1[1] = (tile_w & 0xFFFFu) << 16;                         // tensor_dim0 lo
  g1[2] = ((tile_w >> 16) & 0xFFFFu) | ((tile_h & 0xFFFFu) << 16); // d0 hi|d1 lo
  g1[3] = ((tile_h >> 16) & 0xFFFFu) | ((tile_w & 0xFFFFu) << 16); // d1 hi|tile_dim0
  g1[4] = (tile_h & 0xFFFFu);                               // tile_dim1, dim2=0
  g1[5] = stride_elems;                                     // dim0_stride[31:0]
  g1[6] = 0u;
  g1[7] = 0u;
  asm volatile("tensor_load_to_lds %0, %1" :: "s"(g0), "s"(g1) : "memory");
}

__device__ __forceinline__ void tdm_wait() {
  __builtin_amdgcn_s_wait_tensorcnt(0);
}

// ---------------------------------------------------------------------------
// Kernel 1: qkv = x @ W_qkv^T + b_qkv, scattered into q/k/v [B,H,S,d].
// Block: 16 M-rows x 128 N-cols, 256 threads = 8 waves, 1 16x16 tile per wave.
// Staging: per-wave TDM tiles (x: 2 rows/wave, W: 16 rows/wave).
// ---------------------------------------------------------------------------
__global__ __launch_bounds__(256)
void mha_qkv_kernel(const float* __restrict__ x, const float* __restrict__ W,
                    const float* __restrict__ bias,
                    float* __restrict__ q, float* __restrict__ k,
                    float* __restrict__ v) {
  __shared__ float xs[16][66];
  __shared__ float ws[128][66];
  const int tid  = threadIdx.x;
  const int wave = tid >> 5, lane = tid & 31;
  const int lhi = lane >> 4, llo = lane & 15;
  const int m_base  = blockIdx.x * 16;
  const int n_block = blockIdx.y * 128;

  v8f acc = {0.f, 0.f, 0.f, 0.f, 0.f, 0.f, 0.f, 0.f};

  for (int k0 = 0; k0 < EMB; k0 += 64) {
    __syncthreads();
    // TDM-stage x tile (16x64): wave loads rows [wave*2, wave*2+2)
    tdm_load_tile_pad2((u32)(uintptr_t)&xs[wave * 2][0],
                       x + (size_t)(m_base + wave * 2) * EMB + k0,
                       64, 2, EMB);
    // TDM-stage W tile (128x64): wave loads rows [wave*16, wave*16+16)
    tdm_load_tile_pad2((u32)(uintptr_t)&ws[wave * 16][0],
                       W + (size_t)(n_block + wave * 16) * EMB + k0,
                       64, 16, EMB);
    tdm_wait();
    __syncthreads();
    const int wrow = wave * 16 + llo;
    #pragma unroll
    for (int kk = 0; kk < 64; kk += 4) {
      const int kc = kk + 2 * lhi;
      v2f a = *reinterpret_cast<const v2f*>(&xs[llo][kc]);
      v2f b = *reinterpret_cast<const v2f*>(&ws[wrow][kc]);
      acc = wmma4(a, b, acc);
    }
  }

  // Epilogue: bias + scatter into q/k/v [B,H,S,d]
  const int n_glob = n_block + wave * 16 + llo;
  const float bv = bias[n_glob];
  const int head  = n_glob / 192;       // 3*d = 192 per head
  const int rem   = n_glob % 192;
  const int which = rem / 64;           // 0=q 1=k 2=v
  const int dim   = rem % 64;
  float* dst = (which == 0) ? q : (which == 1) ? k : v;
  const int bidx = m_base >> 11;        // / S_LEN
  const int s0   = m_base & (S_LEN - 1);
  const size_t base = (((size_t)bidx * NH + head) * S_LEN) * HD + dim;
  #pragma unroll
  for (int r = 0; r < 8; ++r) {
    const int m = r + 8 * lhi;
    dst[base + (size_t)(s0 + m) * HD] = acc[r] + bv;
  }
}

// ---------------------------------------------------------------------------
// Kernel 2: attention for one (b,h) pair and 16 query rows.
// 128 threads = 4 waves. Scores (16x2048) live entirely in LDS; double
// softmax in-place; P@V staged through LDS via TDM in 64-key chunks.
// ---------------------------------------------------------------------------
__global__ __launch_bounds__(128)
void mha_attn_kernel(const float* __restrict__ q, const float* __restrict__ k,
                     const float* __restrict__ v, float* __restrict__ o) {
  __shared__ float qs[16][66];          //  ~4.2 KB
  __shared__ float sc[16][2050];        // ~131 KB score stripe
  __shared__ float vs[64][66];          // ~16.9 KB V chunk
  const int tid  = threadIdx.x;
  const int wave = tid >> 5, lane = tid & 31;
  const int lhi = lane >> 4, llo = lane & 15;
  const int bh = blockIdx.x >> 7;             // / (S/16)
  const int q0 = (blockIdx.x & 127) * 16;
  const float* qp = q + (size_t)bh * S_LEN * HD;
  const float* kp = k + (size_t)bh * S_LEN * HD;
  const float* vp = v + (size_t)bh * S_LEN * HD;

  // TDM-stage q tile (16x64): wave loads rows [wave*4, wave*4+4)
  tdm_load_tile_pad2((u32)(uintptr_t)&qs[wave * 4][0],
                     qp + (size_t)(q0 + wave * 4) * HD, 64, 4, HD);
  tdm_wait();
  __syncthreads();

  // ---- scores: sc[m][key] = sqrt(512) * (q . k) ; wave handles 512 keys ----
  const float scale = 22.62741699796952f;     // reference divides by E^-0.5
  for (int kt = 0; kt < 32; ++kt) {
    const int key = wave * 512 + kt * 16 + llo;
    const float* krow = kp + (size_t)key * HD + 2 * lhi;
    if (kt + 1 < 32)                          // global_prefetch_b8 next tile
      __builtin_prefetch(kp + (size_t)(key + 16) * HD, 0, 0);
    v8f acc = {0.f, 0.f, 0.f, 0.f, 0.f, 0.f, 0.f, 0.f};
    #pragma unroll
    for (int kk = 0; kk < 64; kk += 4) {
      v2f a = *reinterpret_cast<const v2f*>(&qs[llo][kk + 2 * lhi]);
      v2f b = *reinterpret_cast<const v2f*>(krow + kk);
      acc = wmma4(a, b, acc);
    }
    #pragma unroll
    for (int r = 0; r < 8; ++r)
      sc[r + 8 * lhi][key] = acc[r] * scale;
  }
  __syncthreads();

  // ---- double softmax over each row; wave handles 4 rows ----
  for (int ri = 0; ri < 4; ++ri) {
    const int row = wave * 4 + ri;
    float mx = -3.4e38f;
    for (int c = lane; c < S_LEN; c += 32) mx = fmaxf(mx, sc[row][c]);
    #pragma unroll
    for (int off = 16; off > 0; off >>= 1)
      mx = fmaxf(mx, __shfl_xor(mx, off, 32));
    float zs = 0.f;
    for (int c = lane; c < S_LEN; c += 32) {
      float e = __expf(sc[row][c] - mx);
      sc[row][c] = e;                   // unnormalized p*Z
      zs += e;
    }
    #pragma unroll
    for (int off = 16; off > 0; off >>= 1) zs += __shfl_xor(zs, off, 32);
    const float invz = 1.f / zs;        // p = e*invz ; max(p) = invz
    float z2 = 0.f;
    for (int c = lane; c < S_LEN; c += 32)
      z2 += __expf(sc[row][c] * invz - invz);
    #pragma unroll
    for (int off = 16; off > 0; off >>= 1) z2 += __shfl_xor(z2, off, 32);
    const float invz2 = 1.f / z2;
    for (int c = lane; c < S_LEN; c += 32)
      sc[row][c] = __expf(sc[row][c] * invz - invz) * invz2;
  }

  // ---- o[16x64] = attn[16x2048] @ V[2048x64]; wave owns 16 dims ----
  v8f acc = {0.f, 0.f, 0.f, 0.f, 0.f, 0.f, 0.f, 0.f};
  for (int ch = 0; ch < 32; ++ch) {
    const int key0 = ch * 64;
    __syncthreads();   // previous chunk consumed (and softmax done, ch==0)
    // TDM-stage V chunk (64x64): wave loads rows [wave*16, wave*16+16)
    tdm_load_tile_pad2((u32)(uintptr_t)&vs[wave * 16][0],
                       vp + (size_t)(key0 + wave * 16) * HD, 64, 16, HD);
    tdm_wait();
    __syncthreads();
    #pragma unroll
    for (int kk = 0; kk < 64; kk += 4) {
      const int kr = kk + 2 * lhi;
      v2f a = *reinterpret_cast<const v2f*>(&sc[llo][key0 + kr]);
      v2f b;
      b.x = vs[kr + 0][wave * 16 + llo];
      b.y = vs[kr + 1][wave * 16 + llo];
      acc = wmma4(a, b, acc);
    }
  }

  // store o to [B,S,E] layout for the projection GEMM
  const int b = bh >> 3, h = bh & 7;
  const int col = h * 64 + wave * 16 + llo;
  #pragma unroll
  for (int r = 0; r < 8; ++r) {
    const int m = r + 8 * lhi;
    o[((size_t)(b * S_LEN + q0 + m)) * EMB + col] = acc[r];
  }
}

// ---------------------------------------------------------------------------
// Kernel 3: out = o @ W_out^T + b_out   (8192x512 @ 512x512)
// ---------------------------------------------------------------------------
__global__ __launch_bounds__(256)
void mha_proj_kernel(const float* __restrict__ xin, const float* __restrict__ W,
                     const float* __restrict__ bias, float* __restrict__ out) {
  __shared__ float xs[16][66];
  __shared__ float ws[128][66];
  const int tid  = threadIdx.x;
  const int wave = tid >> 5, lane = tid & 31;
  const int lhi = lane >> 4, llo = lane & 15;
  const int m_base  = blockIdx.x * 16;
  const int n_block = blockIdx.y * 128;

  v8f acc = {0.f, 0.f, 0.f, 0.f, 0.f, 0.f, 0.f, 0.f};

  for (int k0 = 0; k0 < EMB; k0 += 64) {
    __syncthreads();
    tdm_load_tile_pad2((u32)(uintptr_t)&xs[wave * 2][0],
                       xin + (size_t)(m_base + wave * 2) * EMB + k0,
                       64, 2, EMB);
    tdm_load_tile_pad2((u32)(uintptr_t)&ws[wave * 16][0],
                       W + (size_t)(n_block + wave * 16) * EMB + k0,
                       64, 16, EMB);
    tdm_wait();
    __syncthreads();
    const int wrow = wave * 16 + llo;
    #pragma unroll
    for (int kk = 0; kk < 64; kk += 4) {
      const int kc = kk + 2 * lhi;
      v2f a = *reinterpret_cast<const v2f*>(&xs[llo][kc]);
      v2f b = *reinterpret_cast<const v2f*>(&ws[wrow][kc]);
      acc = wmma4(a, b, acc);
    }
  }

  const int n_glob = n_block + wave * 16 + llo;
  const float bv = bias[n_glob];
  #pragma unroll
  for (int r = 0; r < 8; ++r) {
    const int m = r + 8 * lhi;
    out[(size_t)(m_base + m) * EMB + n_glob] = acc[r] + bv;
  }
}

// ---------------------------------------------------------------------------
extern "C" void kernel_launch(void* const* d_in, const int* in_sizes, int n_in,
                              void* d_out, int out_size, void* d_ws,
                              size_t ws_size, hipStream_t stream) {
  const float* x    = (const float*)d_in[0];
  const float* Wqkv = (const float*)d_in[1];
  const float* bqkv = (const float*)d_in[2];
  const float* Wout = (const float*)d_in[3];
  const float* bout = (const float*)d_in[4];
  float* out = (float*)d_out;

  const size_t tsz = (size_t)BATCH * NH * S_LEN * HD;  // 4,194,304 floats
  float* q = (float*)d_ws;
  float* k = q + tsz;
  float* v = k + tsz;
  float* o = v + tsz;   // o in [B,S,E] layout, same element count

  // qkv projection: M=8192 (512 tiles) x N=1536 (12 blocks of 128)
  mha_qkv_kernel<<<dim3(512, 12), 256, 0, stream>>>(x, Wqkv, bqkv, q, k, v);
  // attention: B*H*(S/16) = 4096 blocks
  mha_attn_kernel<<<dim3(4096), 128, 0, stream>>>(q, k, v, o);
  // output projection: M=8192 x N=512 (4 blocks of 128)
  mha_proj_kernel<<<dim3(512, 4), 256, 0, stream>>>(o, Wout, bout, out);
}